// PercolationQ_31885837205970
// MI455X (gfx1250) — compile-verified
//
#include <hip/hip_runtime.h>

// PercolationQ on MI455X (gfx1250, wave32).
// Bandwidth-bound: ~151 MB fp32 in, 576 floats out -> ~6.5 us at 23.3 TB/s.
// Patch sums are computed on the matrix pipe with V_WMMA_F32_16X16X4_F32
// (A = 16 patches x 4 elems, B = ones, C accumulates) -- exact fp32, so the
// >= 0.59275 threshold compare matches the reference (division by b*b is a
// power of two and thus exact; compare sum >= thr*b*b instead).

typedef __attribute__((ext_vector_type(2))) float v2f;
typedef __attribute__((ext_vector_type(8))) float v8f;

__global__ void zero_out_kernel(float* __restrict__ out, int n) {
    int i = blockIdx.x * blockDim.x + threadIdx.x;
    if (i < n) out[i] = 0.0f;
}

// P               : elements per patch (b*b)
// PATCHES_PER_SLICE: patches per (color, batch) slice
// BLOCKS_PER_SLICE : how many blocks split one slice
template <int P, int PATCHES_PER_SLICE, int BLOCKS_PER_SLICE>
__global__ __launch_bounds__(256) void percolation_q_kernel(
    const float* __restrict__ src, float* __restrict__ out, float thresh) {
    constexpr int PPB = PATCHES_PER_SLICE / BLOCKS_PER_SLICE;  // patches/block
    static_assert(PPB % 128 == 0 || PPB == 128 || PPB == 256 || PPB == 512, "");

    const int slice = blockIdx.x / BLOCKS_PER_SLICE;  // 0..191 == c*64 + b
    const int chunk = blockIdx.x % BLOCKS_PER_SLICE;
    const int wave  = threadIdx.x >> 5;   // 8 waves per block
    const int lane  = threadIdx.x & 31;   // wave32

    const float* sliceBase = src + (size_t)slice * PATCHES_PER_SLICE * P;

    // A-matrix f32 16x4 layout (ISA 7.12.2):
    //   lanes 0-15 : row m=lane,    VGPR0=K(4i+0), VGPR1=K(4i+1)
    //   lanes 16-31: row m=lane-16, VGPR0=K(4i+2), VGPR1=K(4i+3)
    const int row  = (lane < 16) ? lane : (lane - 16);
    const int koff = (lane < 16) ? 0 : 2;

    v2f ones;
    ones.x = 1.0f;
    ones.y = 1.0f;

    float cnt = 0.0f;  // per-thread count of above-threshold patches (x16 dup)

    const int pBegin = chunk * PPB;
    const int pEnd   = pBegin + PPB;
    // Uniform loop per wave: EXEC stays all-1s through every WMMA.
    for (int p0 = pBegin + wave * 16; p0 < pEnd; p0 += 8 * 16) {
        const float* base = sliceBase + (size_t)(p0 + row) * P + koff;
        v8f acc = {};
#pragma unroll
        for (int i = 0; i < P / 4; ++i) {
            v2f a = *(const v2f*)(base + 4 * i);  // global_load_b64
            // D = A x ones + C : every column of D = per-patch partial sum
            acc = __builtin_amdgcn_wmma_f32_16x16x4_f32(
                /*neg_a=*/false, a, /*neg_b=*/false, ones,
                /*c_mod=*/(short)0, acc, /*reuse_a=*/false, /*reuse_b=*/false);
        }
        // D VGPR j: lanes 0-15 -> patchsum[p0+j], lanes 16-31 -> patchsum[p0+8+j]
#pragma unroll
        for (int j = 0; j < 8; ++j)
            cnt += (acc[j] >= thresh) ? 1.0f : 0.0f;
    }

    // Block reduction; every patch was counted 16x (16 identical D columns).
    __shared__ float red[256];
    red[threadIdx.x] = cnt;
    __syncthreads();
#pragma unroll
    for (int s = 128; s > 0; s >>= 1) {
        if (threadIdx.x < s) red[threadIdx.x] += red[threadIdx.x + s];
        __syncthreads();
    }
    if (threadIdx.x == 0) {
        atomicAdd(&out[slice],
                  red[0] * (1.0f / 16.0f) * (1.0f / (float)PATCHES_PER_SLICE));
    }
}

extern "C" void kernel_launch(void* const* d_in, const int* in_sizes, int n_in,
                              void* d_out, int out_size, void* d_ws, size_t ws_size,
                              hipStream_t stream) {
    const float* x4  = (const float*)d_in[0];  // [3,64,4096,4,4]
    const float* x8  = (const float*)d_in[1];  // [3,64,1024,8,8]
    const float* x16 = (const float*)d_in[2];  // [3,64,256,16,16]
    float* out = (float*)d_out;                // q4[192] | q8[192] | q16[192]

    zero_out_kernel<<<(out_size + 255) / 256, 256, 0, stream>>>(out, out_size);

    const float T = 0.59275f;
    // thresh on raw sum: T * b*b (power-of-two scale -> exact in fp32)
    percolation_q_kernel<16, 4096, 8>
        <<<192 * 8, 256, 0, stream>>>(x4, out + 0, T * 16.0f);
    percolation_q_kernel<64, 1024, 2>
        <<<192 * 2, 256, 0, stream>>>(x8, out + 192, T * 64.0f);
    percolation_q_kernel<256, 256, 1>
        <<<192 * 1, 256, 0, stream>>>(x16, out + 384, T * 256.0f);
}